// CodeBook_56435870270008
// MI455X (gfx1250) — compile-verified
//
#include <hip/hip_runtime.h>
#include <stdint.h>

#define LEVELS    8
#define DICT_SIZE 256
#define FEAT_DIM  128
#define NUM_WORDS 100000
#define N_TOT     100000

#define BLOCK_T   256
#define GRID_X    256

typedef float f4  __attribute__((ext_vector_type(4)));
typedef int   v4i __attribute__((ext_vector_type(4)));

// Pointer-to-AS1(int4) and pointer-to-AS3(int4), matching the async builtin's params.
typedef __attribute__((address_space(1))) v4i* gv4i_p;
typedef __attribute__((address_space(3))) v4i* lv4i_p;

#if defined(__AMDGCN__) && __has_builtin(__builtin_amdgcn_global_load_async_to_lds_b128)
#define HAVE_ASYNC_LDS 1
#else
#define HAVE_ASYNC_LDS 0
#endif

__global__ __launch_bounds__(BLOCK_T)
void codebook_argmax_gather_kernel(const int*   __restrict__ idx,
                                   const float* __restrict__ dict,
                                   const float* __restrict__ feats,
                                   float*       __restrict__ out)
{
    // One level's dictionary: 256 x 128 f32 = 128 KB of CDNA5's 320 KB LDS.
    __shared__ __align__(16) float s_dict[DICT_SIZE * FEAT_DIM];

    const int l   = blockIdx.y;        // level 0..7
    const int tid = threadIdx.x;

    const float* dlvl = dict + (size_t)l * (DICT_SIZE * FEAT_DIM);

#if HAVE_ASYNC_LDS
    // Stage dictionary[l] into LDS with async global->LDS b128 copies (ASYNCcnt).
    for (int i = tid; i < DICT_SIZE * FEAT_DIM / 4; i += BLOCK_T) {
        __builtin_amdgcn_global_load_async_to_lds_b128(
            (gv4i_p)(dlvl + 4 * i),
            (lv4i_p)(&s_dict[4 * i]),
            0, 0);
    }
#if __has_builtin(__builtin_amdgcn_s_wait_asynccnt)
    __builtin_amdgcn_s_wait_asynccnt(0);
#else
    asm volatile("s_wait_asynccnt 0" ::: "memory");
#endif
#else
    // Fallback: regular b128 load + ds b128 store staging.
    for (int i = tid; i < DICT_SIZE * FEAT_DIM / 4; i += BLOCK_T) {
        ((f4*)s_dict)[i] = ((const f4*)dlvl)[i];
    }
#endif
    __syncthreads();

    const int lane  = tid & 31;
    const int wave  = tid >> 5;
    const int gwave = blockIdx.x * (BLOCK_T / 32) + wave;
    const int nwv   = GRID_X * (BLOCK_T / 32);

    const f4* feats4 = (const f4*)(feats + (size_t)l * NUM_WORDS * DICT_SIZE);
    const f4* sdict4 = (const f4*)s_dict;
    f4*       out4   = (f4*)out;

    for (int n = gwave; n < N_TOT; n += nwv) {
        const int w = __builtin_amdgcn_readfirstlane(idx[n]);
        const f4* row = feats4 + (size_t)w * (DICT_SIZE / 4);

        // Prefetch the row two iterations ahead (global_prefetch_b8):
        // lanes 0..7 at 128B stride cover the whole 1KB row.
        const int n2 = n + 2 * nwv;
        if (n2 < N_TOT) {
            const int w2 = __builtin_amdgcn_readfirstlane(idx[n2]);
            __builtin_prefetch((const void*)(feats4 + (size_t)w2 * (DICT_SIZE / 4)
                                             + (size_t)((lane & 7) * 8)), 0, 0);
        }

        // Whole wave reads the 256-float row: 2 coalesced non-temporal
        // b128 loads per lane (feats is streamed, ~zero reuse).
        const f4 a = __builtin_nontemporal_load(row + lane);        // k = 4*lane + j
        const f4 b = __builtin_nontemporal_load(row + lane + 32);   // k = 128 + 4*lane + j

        // Lane-local argmax with first-occurrence tie-break (strict >,
        // scanned in ascending-k order within this lane's elements).
        float bv = a.x; int bk = 4 * lane;
        if (a.y > bv) { bv = a.y; bk = 4 * lane + 1; }
        if (a.z > bv) { bv = a.z; bk = 4 * lane + 2; }
        if (a.w > bv) { bv = a.w; bk = 4 * lane + 3; }
        if (b.x > bv) { bv = b.x; bk = 128 + 4 * lane; }
        if (b.y > bv) { bv = b.y; bk = 128 + 4 * lane + 1; }
        if (b.z > bv) { bv = b.z; bk = 128 + 4 * lane + 2; }
        if (b.w > bv) { bv = b.w; bk = 128 + 4 * lane + 3; }

        // Pack (order-preserving float encoding, 255-k) so a u64 max gives
        // the max value and, on ties, the smallest index k.
        unsigned u = __float_as_uint(bv);
        u = (u & 0x80000000u) ? ~u : (u | 0x80000000u);
        unsigned long long key =
            ((unsigned long long)u << 32) | (unsigned)(255 - bk);

        // wave32 butterfly reduction (5 steps).
        for (int off = 16; off > 0; off >>= 1) {
            unsigned long long o = __shfl_xor(key, off, 32);
            if (o > key) key = o;
        }
        const int code = 255 - (int)(key & 0xFFull);

        // Gather codeword from LDS (ds_load_b128) and stream out the result:
        // 512B contiguous per wave, non-temporal (write-once output).
        const f4 cw = sdict4[code * (FEAT_DIM / 4) + lane];
        __builtin_nontemporal_store(
            cw, out4 + (size_t)n * (LEVELS * FEAT_DIM / 4) + l * (FEAT_DIM / 4) + lane);
    }
}

extern "C" void kernel_launch(void* const* d_in, const int* in_sizes, int n_in,
                              void* d_out, int out_size, void* d_ws, size_t ws_size,
                              hipStream_t stream) {
    (void)in_sizes; (void)n_in; (void)out_size; (void)d_ws; (void)ws_size;

    const int*   idx   = (const int*)  d_in[0];  // [N] int32
    const float* dict  = (const float*)d_in[1];  // [8, 256, 128] f32
    const float* feats = (const float*)d_in[2];  // [8, 100000, 256] f32
    float*       out   = (float*)      d_out;    // [N, 1024] f32

    dim3 grid(GRID_X, LEVELS, 1);
    codebook_argmax_gather_kernel<<<grid, BLOCK_T, 0, stream>>>(idx, dict, feats, out);
}